// VoltageViolationLoss_9603546874515
// MI455X (gfx1250) — compile-verified
//
#include <hip/hip_runtime.h>
#include <hip/hip_bf16.h>
#include <math.h>

// ---------------- problem constants ----------------
#define BATCH    512
#define NREAL    1023          // N buses
#define NPAD     1024          // padded bus dimension
#define NUM_CS   256
#define STATE_W  2818          // 4 + 2*1023 + 3*256
#define EV_START 2050          // 4 + 2*1023
#define ITERS    100
#define TOL      1e-6f

typedef __attribute__((ext_vector_type(2))) float v2f;
typedef __attribute__((ext_vector_type(4))) float v4f;
typedef __attribute__((ext_vector_type(8))) float v8f;

// ---------------- workspace layout (bytes) ----------------
static constexpr size_t OFF_KTR  = 0;
static constexpr size_t OFF_KTI  = OFF_KTR + (size_t)NPAD * NPAD * 4;
static constexpr size_t OFF_SR   = OFF_KTI + (size_t)NPAD * NPAD * 4;
static constexpr size_t OFF_SI   = OFF_SR  + (size_t)BATCH * NPAD * 4;
static constexpr size_t OFF_VAR  = OFF_SI  + (size_t)BATCH * NPAD * 4;
static constexpr size_t OFF_VAI  = OFF_VAR + (size_t)BATCH * NPAD * 4;
static constexpr size_t OFF_VBR  = OFF_VAI + (size_t)BATCH * NPAD * 4;
static constexpr size_t OFF_VBI  = OFF_VBR + (size_t)BATCH * NPAD * 4;
static constexpr size_t OFF_LTR  = OFF_VBI + (size_t)BATCH * NPAD * 4;
static constexpr size_t OFF_LTI  = OFF_LTR + (size_t)BATCH * NPAD * 4;
static constexpr size_t OFF_CTRL = OFF_LTI + (size_t)BATCH * NPAD * 4;
static constexpr size_t WS_NEED  = OFF_CTRL + 256;

// ---------------- transpose + pad K into Kt (Kt[j][i] = K[i][j]) -------------
__global__ __launch_bounds__(256)
void k_transpose_pad(const float* __restrict__ src, float* __restrict__ dst) {
    __shared__ float tile[32][33];
    const int bx = blockIdx.x * 32;     // src row base (i)
    const int by = blockIdx.y * 32;     // src col base (j)
    const int tx = threadIdx.x & 31;
    const int ty = threadIdx.x >> 5;    // 0..7
    #pragma unroll
    for (int rr = 0; rr < 32; rr += 8) {
        const int i = bx + ty + rr;
        const int j = by + tx;
        tile[ty + rr][tx] = (i < NREAL && j < NREAL) ? src[(size_t)i * NREAL + j] : 0.0f;
    }
    __syncthreads();
    #pragma unroll
    for (int rr = 0; rr < 32; rr += 8) {
        const int j = by + ty + rr;     // dst row
        const int i = bx + tx;          // dst col
        dst[(size_t)j * NPAD + i] = tile[tx][ty + rr];
    }
}

// ---------------- init: S_r = 0, v = 1+0j, ctrl init -------------------------
__global__ __launch_bounds__(256)
void k_init(float* __restrict__ Sr, float* __restrict__ vr, float* __restrict__ vi,
            float* __restrict__ ctrl) {
    const int i = blockIdx.x * 256 + threadIdx.x;   // 0 .. 512*1024-1
    Sr[i] = 0.0f;
    vr[i] = 1.0f;
    vi[i] = 0.0f;
    if (i == 0) {
        ctrl[0] = __builtin_inff();                 // tol
        ((int*)ctrl)[1] = 1;                        // flag (active)
        ((unsigned int*)ctrl)[2] = 0u;              // tol_acc
    }
}

// ---------------- EV power scatter into S_r ---------------------------------
__global__ __launch_bounds__(256)
void k_scatter(const float* __restrict__ action, const float* __restrict__ state,
               float* __restrict__ Sr) {
    const int t = blockIdx.x * 256 + threadIdx.x;   // 0 .. 512*256-1
    const int b = t >> 8;
    const int c = t & 255;
    const float* srow = state + (size_t)b * STATE_W;
    const float cap  = srow[EV_START + 3 * c];
    const float busf = srow[EV_START + 3 * c + 2];
    const float conn = (cap > 0.0f) ? 1.0f : 0.0f;
    const float maxc = fminf(22.0f,  conn * (70.0f - cap) * 4.0f);   // /TIMESCALE
    const float maxd = fmaxf(-22.0f, conn * (15.0f - cap) * 4.0f);
    const float a = action[t];
    // a*MAX*bin - a*MIN*(1-bin) collapses to 22.17*a for both signs
    float pu = 22.17f * a;
    pu = fminf(pu, maxc);
    pu = fmaxf(pu, maxd);
    const int bus = (int)busf;                      // 0 .. NREAL-1
    atomicAdd(&Sr[(size_t)b * NPAD + bus], pu);
}

// ---------------- finalize S: S_r=(active+P)/1000, S_i=reactive/1000 --------
__global__ __launch_bounds__(256)
void k_finalize_S(const float* __restrict__ state, float* __restrict__ Sr,
                  float* __restrict__ Si) {
    const int i = blockIdx.x * 256 + threadIdx.x;
    const int b = i >> 10;
    const int j = i & (NPAD - 1);
    if (j < NREAL) {
        const float* srow = state + (size_t)b * STATE_W;
        Sr[i] = (srow[4 + j] + Sr[i]) * 1e-3f;
        Si[i] = srow[4 + NREAL + j] * 1e-3f;
    } else {
        Sr[i] = 0.0f;
        Si[i] = 0.0f;
    }
}

// ---------------- L_term = conj(S) * v / |v|^2 ------------------------------
__global__ __launch_bounds__(256)
void k_compute_L(const float* __restrict__ vr_, const float* __restrict__ vi_,
                 const float* __restrict__ Sr, const float* __restrict__ Si,
                 float* __restrict__ Lr, float* __restrict__ Li) {
    const int i = blockIdx.x * 256 + threadIdx.x;
    const int j = i & (NPAD - 1);
    if (j >= NREAL) { Lr[i] = 0.0f; Li[i] = 0.0f; return; }
    const float vr = vr_[i], vi = vi_[i];
    const float sr = Sr[i],  si = Si[i];
    const float inv = 1.0f / (vr * vr + vi * vi);
    Lr[i] = (sr * vr + si * vi) * inv;
    Li[i] = (sr * vi - si * vr) * inv;
}

// ---------------- core: v_next = L @ K^T + W (complex), fp32 WMMA -----------
// block = 256 threads (8 waves, 2x4 wave grid), 64(M) x 64(N) block tile,
// BK = 32. Each wave owns a 32x16 strip = two 16x16 WMMA tiles stacked in M,
// so B fragments are reused across both tiles: 8 WMMA per 6 DS loads.
// When the reference's scan-carry flag is frozen, the result of the GEMM is
// discarded by the reference, so we uniformly skip it and copy v instead.
__global__ __launch_bounds__(256)
void k_pf_gemm(const float* __restrict__ Lr,  const float* __restrict__ Li,
               const float* __restrict__ Ktr, const float* __restrict__ Kti,
               const float* __restrict__ vcr, const float* __restrict__ vci,
               float* __restrict__ vnr, float* __restrict__ vni,
               const float* __restrict__ Wr_in, const float* __restrict__ Wi_in,
               float* __restrict__ ctrl) {
    __shared__ float As_r[64][33];
    __shared__ float As_i[64][33];
    __shared__ float Bs_r[32][65];
    __shared__ float Bs_i[32][65];

    const int tid   = threadIdx.x;
    const int tileN = blockIdx.x * 64;
    const int tileM = blockIdx.y * 64;

    // ---- uniform frozen check: skip all compute, copy v ----
    const int frozen = (((const int*)ctrl)[1] == 0);
    if (frozen) {
        // copy 64x64 tile: thread t -> row t>>2, 16-col chunk (t&3)*16
        const int m  = tid >> 2;
        const int c0 = (tid & 3) * 16;
        const size_t base = (size_t)(tileM + m) * NPAD + tileN + c0;
        #pragma unroll
        for (int q = 0; q < 16; q += 4) {
            *(v4f*)(vnr + base + q) = *(const v4f*)(vcr + base + q);
            *(v4f*)(vni + base + q) = *(const v4f*)(vci + base + q);
        }
        return;
    }

    const int lane = tid & 31;
    const int wave = tid >> 5;
    const int wm2  = wave >> 2;        // 0..1 : 32-row strip
    const int wn   = wave & 3;         // 0..3 : 16-col strip
    const int half = lane >> 4;        // 0/1
    const int lm   = lane & 15;

    v8f acc0_r = {}, acc0_i = {};      // rows wm2*32      .. +15
    v8f acc1_r = {}, acc1_i = {};      // rows wm2*32 + 16 .. +31

    for (int kk = 0; kk < NPAD; kk += 32) {
        __syncthreads();
        // stage A (L_term) 64x32 tile: 4 x v2f per part per thread
        #pragma unroll
        for (int p = 0; p < 4; ++p) {
            const int e2 = tid + 256 * p;        // 0..1023 (pairs)
            const int m  = e2 >> 4;              // 0..63
            const int k  = (e2 & 15) * 2;        // 0..30
            const size_t ga = (size_t)(tileM + m) * NPAD + kk + k;
            const v2f ar = *(const v2f*)(Lr + ga);
            const v2f ai = *(const v2f*)(Li + ga);
            As_r[m][k] = ar.x; As_r[m][k + 1] = ar.y;
            As_i[m][k] = ai.x; As_i[m][k + 1] = ai.y;
        }
        // stage B (K^T) 32x64 tile
        #pragma unroll
        for (int p = 0; p < 4; ++p) {
            const int e2 = tid + 256 * p;
            const int k  = e2 >> 5;              // 0..31
            const int n  = (e2 & 31) * 2;        // 0..62
            const size_t gb = (size_t)(kk + k) * NPAD + tileN + n;
            const v2f br = *(const v2f*)(Ktr + gb);
            const v2f bi = *(const v2f*)(Kti + gb);
            Bs_r[k][n] = br.x; Bs_r[k][n + 1] = br.y;
            Bs_i[k][n] = bi.x; Bs_i[k][n + 1] = bi.y;
        }
        __syncthreads();

        const int arow0 = wm2 * 32 + lm;
        const int arow1 = arow0 + 16;
        const int bcol  = wn * 16 + lm;
        #pragma unroll
        for (int k4 = 0; k4 < 8; ++k4) {
            const int k0 = k4 * 4;
            v2f a0r, a0i, a0in, a1r, a1i, a1in, br, bi;
            // A 16x4 frag: lanes 0-15 hold K=k0,k0+1 ; lanes 16-31 hold K=k0+2,k0+3
            a0r.x = As_r[arow0][k0 + 2 * half];
            a0r.y = As_r[arow0][k0 + 2 * half + 1];
            a0i.x = As_i[arow0][k0 + 2 * half];
            a0i.y = As_i[arow0][k0 + 2 * half + 1];
            a1r.x = As_r[arow1][k0 + 2 * half];
            a1r.y = As_r[arow1][k0 + 2 * half + 1];
            a1i.x = As_i[arow1][k0 + 2 * half];
            a1i.y = As_i[arow1][k0 + 2 * half + 1];
            a0in = -a0i;   // f32 WMMA has no A/B NEG modifier -> negate in VALU
            a1in = -a1i;
            // B 4x16 frag: vgpr0 = rows k0/k0+1 across lane halves, vgpr1 = k0+2/k0+3
            br.x = Bs_r[k0 + half][bcol];
            br.y = Bs_r[k0 + 2 + half][bcol];
            bi.x = Bs_i[k0 + half][bcol];
            bi.y = Bs_i[k0 + 2 + half][bcol];
            // acc_r += ar*br - ai*bi ; acc_i += ar*bi + ai*br   (B reused 2x)
            acc0_r = __builtin_amdgcn_wmma_f32_16x16x4_f32(false, a0r,  false, br, (short)0, acc0_r, false, false);
            acc0_r = __builtin_amdgcn_wmma_f32_16x16x4_f32(false, a0in, false, bi, (short)0, acc0_r, false, false);
            acc0_i = __builtin_amdgcn_wmma_f32_16x16x4_f32(false, a0r,  false, bi, (short)0, acc0_i, false, false);
            acc0_i = __builtin_amdgcn_wmma_f32_16x16x4_f32(false, a0i,  false, br, (short)0, acc0_i, false, false);
            acc1_r = __builtin_amdgcn_wmma_f32_16x16x4_f32(false, a1r,  false, br, (short)0, acc1_r, false, false);
            acc1_r = __builtin_amdgcn_wmma_f32_16x16x4_f32(false, a1in, false, bi, (short)0, acc1_r, false, false);
            acc1_i = __builtin_amdgcn_wmma_f32_16x16x4_f32(false, a1r,  false, bi, (short)0, acc1_i, false, false);
            acc1_i = __builtin_amdgcn_wmma_f32_16x16x4_f32(false, a1i,  false, br, (short)0, acc1_i, false, false);
        }
    }

    // ---- epilogue: +W, tolerance atomic-max (active path only) ----
    const int gn = tileN + wn * 16 + lm;
    const float Wr = (gn < NREAL) ? Wr_in[gn] : 0.0f;
    const float Wi = (gn < NREAL) ? Wi_in[gn] : 0.0f;
    float lmax = 0.0f;
    #pragma unroll
    for (int tsub = 0; tsub < 2; ++tsub) {
        const v8f ar = tsub ? acc1_r : acc0_r;
        const v8f ai = tsub ? acc1_i : acc0_i;
        #pragma unroll
        for (int r = 0; r < 8; ++r) {
            const int gm = tileM + wm2 * 32 + tsub * 16 + r + 8 * half;
            const size_t idx = (size_t)gm * NPAD + gn;
            float vr = ar[r] + Wr;
            float vi = ai[r] + Wi;
            const float ovr = vcr[idx];
            const float ovi = vci[idx];
            if (gn >= NREAL) { vr = 1.0f; vi = 0.0f; }   // keep pad column benign
            vnr[idx] = vr;
            vni[idx] = vi;
            if (gn < NREAL) {
                const float d = fabsf(sqrtf(vr * vr + vi * vi) -
                                      sqrtf(ovr * ovr + ovi * ovi));
                lmax = fmaxf(lmax, d);
            }
        }
    }
    // wave32 max-reduce, one atomic per wave (vals >= 0 -> uint order == float order)
    #pragma unroll
    for (int o = 16; o > 0; o >>= 1) lmax = fmaxf(lmax, __shfl_down(lmax, o, 32));
    if (lane == 0 && lmax > 0.0f)
        atomicMax(((unsigned int*)ctrl) + 2, __float_as_uint(lmax));
}

// ---------------- per-iteration tolerance carry -----------------------------
__global__ void k_tol_update(float* __restrict__ ctrl) {
    int*          flag = ((int*)ctrl) + 1;
    unsigned int* acc  = ((unsigned int*)ctrl) + 2;
    if (*flag) ctrl[0] = __uint_as_float(*acc);
    *acc = 0u;
    *flag = (ctrl[0] >= TOL) ? 1 : 0;
}

// ---------------- loss: 1000 * sum_j min(0, 0.05 - |1 - |v||) ---------------
__global__ __launch_bounds__(256)
void k_loss(const float* __restrict__ vr_, const float* __restrict__ vi_,
            float* __restrict__ out) {
    __shared__ float sred[256];
    const int b = blockIdx.x;
    const int tid = threadIdx.x;
    float s = 0.0f;
    for (int j = tid; j < NREAL; j += 256) {
        const size_t idx = (size_t)b * NPAD + j;
        const float vr = vr_[idx], vi = vi_[idx];
        const float mag = sqrtf(vr * vr + vi * vi);
        s += fminf(0.0f, 0.05f - fabsf(1.0f - mag));
    }
    sred[tid] = s;
    __syncthreads();
    for (int o = 128; o > 0; o >>= 1) {
        if (tid < o) sred[tid] += sred[tid + o];
        __syncthreads();
    }
    if (tid == 0) out[b] = 1000.0f * sred[0];
}

// ---------------- host driver ----------------------------------------------
extern "C" void kernel_launch(void* const* d_in, const int* in_sizes, int n_in,
                              void* d_out, int out_size, void* d_ws, size_t ws_size,
                              hipStream_t stream) {
    if (ws_size < WS_NEED) return;   // workspace too small: nothing safe to do

    const float* action = (const float*)d_in[0];
    const float* state  = (const float*)d_in[1];
    const float* K_r    = (const float*)d_in[2];
    const float* K_i    = (const float*)d_in[3];
    const float* W_r    = (const float*)d_in[4];   // reference L_r
    const float* W_i    = (const float*)d_in[5];   // reference L_i
    float* out = (float*)d_out;

    char* ws = (char*)d_ws;
    float* Ktr  = (float*)(ws + OFF_KTR);
    float* Kti  = (float*)(ws + OFF_KTI);
    float* Sr   = (float*)(ws + OFF_SR);
    float* Si   = (float*)(ws + OFF_SI);
    float* vAr  = (float*)(ws + OFF_VAR);
    float* vAi  = (float*)(ws + OFF_VAI);
    float* vBr  = (float*)(ws + OFF_VBR);
    float* vBi  = (float*)(ws + OFF_VBI);
    float* Ltr  = (float*)(ws + OFF_LTR);
    float* Lti  = (float*)(ws + OFF_LTI);
    float* ctrl = (float*)(ws + OFF_CTRL);

    // once-per-launch preprocessing (deterministic, graph-capture safe)
    k_transpose_pad<<<dim3(32, 32), 256, 0, stream>>>(K_r, Ktr);
    k_transpose_pad<<<dim3(32, 32), 256, 0, stream>>>(K_i, Kti);
    k_init<<<(BATCH * NPAD) / 256, 256, 0, stream>>>(Sr, vAr, vAi, ctrl);
    k_scatter<<<(BATCH * NUM_CS) / 256, 256, 0, stream>>>(action, state, Sr);
    k_finalize_S<<<(BATCH * NPAD) / 256, 256, 0, stream>>>(state, Sr, Si);

    // 100 fixed-point iterations (ping-pong voltage buffers)
    float* cr = vAr; float* ci = vAi;
    float* nr = vBr; float* ni = vBi;
    for (int it = 0; it < ITERS; ++it) {
        k_compute_L<<<(BATCH * NPAD) / 256, 256, 0, stream>>>(cr, ci, Sr, Si, Ltr, Lti);
        k_pf_gemm<<<dim3(NPAD / 64, BATCH / 64), 256, 0, stream>>>(
            Ltr, Lti, Ktr, Kti, cr, ci, nr, ni, W_r, W_i, ctrl);
        k_tol_update<<<1, 1, 0, stream>>>(ctrl);
        float* tr = cr; cr = nr; nr = tr;
        float* ti = ci; ci = ni; ni = ti;
    }

    k_loss<<<BATCH, 256, 0, stream>>>(cr, ci, out);
}